// gasGATConv_52871047413958
// MI455X (gfx1250) — compile-verified
//
#include <hip/hip_runtime.h>
#include <math.h>

// ---- problem constants (match reference) ----
#define IN_CH     256
#define OUT_CH    64
#define NEG_SLOPE 0.2f

typedef float v2f __attribute__((ext_vector_type(2)));
typedef float v8f __attribute__((ext_vector_type(8)));

// Types for the async global->LDS builtin (param type per hipcc diagnostic:
// pointer to int __attribute__((vector_size(16))) in AS1 / AS3)
typedef int v4i_vs __attribute__((vector_size(16)));
typedef __attribute__((address_space(1))) v4i_vs* as1_v4i_ptr;
typedef __attribute__((address_space(3))) v4i_vs* as3_v4i_ptr;

// Probe for gfx1250 async global->LDS builtins (toolchain dependent)
#if defined(__has_builtin)
#  if __has_builtin(__builtin_amdgcn_global_load_async_to_lds_b128)
#    define HAVE_ASYNC_LDS 1
#  endif
#  if __has_builtin(__builtin_amdgcn_s_wait_asynccnt)
#    define HAVE_WAIT_ASYNC 1
#  endif
#endif

// ---------------------------------------------------------------------------
// Kernel 1: feature = x @ W^T + b  via V_WMMA_F32_16X16X4_F32
// Block = 128 threads (4 wave32). Block tile: 16 nodes x 64 channels.
// Wave w handles channels [16w, 16w+16). K-loop: 64 rank-4 WMMA steps.
// x tile staged HBM -> LDS via async-to-LDS b128 when available.
// ---------------------------------------------------------------------------
__global__ void __launch_bounds__(128)
gat_gemm_wmma(const float* __restrict__ x, const float* __restrict__ W,
              const float* __restrict__ b, float* __restrict__ feature,
              int n_nodes) {
    __shared__ float xs[16][IN_CH + 4];   // +4 pad: conflict-free row access
                                          // row stride 1040B (16B aligned)

    const int m0   = blockIdx.x * 16;
    const int tid  = threadIdx.x;
    const int wave = tid >> 5;
    const int lane = tid & 31;
    const int half = lane >> 4;
    const int ln   = lane & 15;

    // ---- stage the 16 x 256 x-tile into LDS ----
#if HAVE_ASYNC_LDS
    for (int i = tid; i < 16 * (IN_CH / 4); i += 128) {
        const int r   = i >> 6;          // / (IN_CH/4)
        const int c4  = i & 63;          // % (IN_CH/4)
        const int row = m0 + r;
        if (row < n_nodes) {
            // direct HBM -> LDS, tracked by ASYNCcnt (no VGPR round-trip)
            __builtin_amdgcn_global_load_async_to_lds_b128(
                (as1_v4i_ptr)(x + (size_t)row * IN_CH + c4 * 4),
                (as3_v4i_ptr)(&xs[r][c4 * 4]),
                /*offset=*/0, /*cpol=*/0);
        } else {
            xs[r][c4 * 4 + 0] = 0.f;
            xs[r][c4 * 4 + 1] = 0.f;
            xs[r][c4 * 4 + 2] = 0.f;
            xs[r][c4 * 4 + 3] = 0.f;
        }
    }
#  if HAVE_WAIT_ASYNC
    __builtin_amdgcn_s_wait_asynccnt(0);
#  else
    asm volatile("s_wait_asynccnt 0x0" ::: "memory");
#  endif
#else
    for (int i = tid; i < 16 * (IN_CH / 4); i += 128) {
        const int r   = i >> 6;
        const int c4  = i & 63;
        const int row = m0 + r;
        float4 v = make_float4(0.f, 0.f, 0.f, 0.f);
        if (row < n_nodes)
            v = ((const float4*)(x + (size_t)row * IN_CH))[c4];
        xs[r][c4 * 4 + 0] = v.x;
        xs[r][c4 * 4 + 1] = v.y;
        xs[r][c4 * 4 + 2] = v.z;
        xs[r][c4 * 4 + 3] = v.w;
    }
#endif
    __syncthreads();

    const int n0 = wave * 16;                 // channel tile base
    const float bias = b[n0 + ln];
    v8f acc = {bias, bias, bias, bias, bias, bias, bias, bias};

    const float* __restrict__ Wrow = W + (size_t)(n0 + ln) * IN_CH;

#pragma unroll 8
    for (int k = 0; k < IN_CH; k += 4) {
        v2f a, bb;
        a[0]  = xs[ln][k + 2 * half + 0];     // A[M=ln][K=k+2h+0]
        a[1]  = xs[ln][k + 2 * half + 1];     // A[M=ln][K=k+2h+1]
        bb[0] = Wrow[k + 2 * half + 0];       // B[K][N=ln] = W[N][K]
        bb[1] = Wrow[k + 2 * half + 1];
        acc = __builtin_amdgcn_wmma_f32_16x16x4_f32(
            /*neg_a=*/false, a, /*neg_b=*/false, bb,
            /*c_mod=*/(short)0, acc, /*reuse_a=*/false, /*reuse_b=*/false);
    }

    // D[r + 8*half][ln] -> feature[m0 + r + 8*half][n0 + ln]
    const int mbase = m0 + 8 * half;
    float* __restrict__ fout = feature + (size_t)mbase * OUT_CH + n0 + ln;
    if (m0 + 16 <= n_nodes) {                 // full tile: unguarded stores
#pragma unroll
        for (int r = 0; r < 8; ++r)
            fout[(size_t)r * OUT_CH] = acc[r];
    } else {
#pragma unroll
        for (int r = 0; r < 8; ++r)
            if (mbase + r < n_nodes)
                fout[(size_t)r * OUT_CH] = acc[r];
    }
}

// ---------------------------------------------------------------------------
// Kernel 2: per-node e_pre = feature @ att  (wave per node, shfl reduction)
//           also init m[n] = logit_self = leakyrelu(e0 + e1)
// ---------------------------------------------------------------------------
__global__ void __launch_bounds__(256)
gat_epre(const float* __restrict__ feature, const float* __restrict__ att,
         float* __restrict__ e_pre, float* __restrict__ m, int n_nodes) {
    const int node = blockIdx.x * 8 + (threadIdx.x >> 5);
    const int lane = threadIdx.x & 31;
    if (node >= n_nodes) return;

    const float* __restrict__ f = feature + (size_t)node * OUT_CH;
    const float f0 = f[lane];
    const float f1 = f[lane + 32];
    float e0 = f0 * att[lane * 2 + 0] + f1 * att[(lane + 32) * 2 + 0];
    float e1 = f0 * att[lane * 2 + 1] + f1 * att[(lane + 32) * 2 + 1];
#pragma unroll
    for (int off = 16; off > 0; off >>= 1) {
        e0 += __shfl_xor(e0, off, 32);
        e1 += __shfl_xor(e1, off, 32);
    }
    if (lane == 0) {
        e_pre[node * 2 + 0] = e0;
        e_pre[node * 2 + 1] = e1;
        float ls = e0 + e1;
        ls = (ls > 0.f) ? ls : ls * NEG_SLOPE;
        m[node] = ls;                 // max starts at the self-loop logit
    }
}

// Ordered float atomic max via sign-aware integer atomics.
__device__ __forceinline__ void atomicMaxFloat(float* addr, float val) {
    if (val >= 0.f)
        atomicMax((int*)addr, __float_as_int(val));
    else
        atomicMin((unsigned int*)addr, __float_as_uint(val));
}

// ---------------------------------------------------------------------------
// Kernel 3: per-edge logit + segment max into m[tar]
// ---------------------------------------------------------------------------
__global__ void __launch_bounds__(256)
gat_edge_logit(const long long* __restrict__ ei, const float* __restrict__ e_pre,
               float* __restrict__ ebuf, float* __restrict__ m, int n_edges) {
    const int e = blockIdx.x * 256 + threadIdx.x;
    if (e >= n_edges) return;
    const long long t = ei[e];
    const long long s = ei[n_edges + e];
    float v = e_pre[t * 2 + 0] + e_pre[s * 2 + 1];
    v = (v > 0.f) ? v : v * NEG_SLOPE;
    ebuf[e] = v;                      // stash logit for the exp pass
    atomicMaxFloat(&m[t], v);
}

// ---------------------------------------------------------------------------
// Kernel 4: per-node denom init with exp_self (also stash exp_self)
// ---------------------------------------------------------------------------
__global__ void __launch_bounds__(256)
gat_denom_init(const float* __restrict__ e_pre, const float* __restrict__ m,
               float* __restrict__ denom, float* __restrict__ exps, int n_nodes) {
    const int n = blockIdx.x * 256 + threadIdx.x;
    if (n >= n_nodes) return;
    float ls = e_pre[n * 2 + 0] + e_pre[n * 2 + 1];
    ls = (ls > 0.f) ? ls : ls * NEG_SLOPE;
    const float es = expf(ls - m[n]);
    denom[n] = es;
    exps[n]  = es;
}

// ---------------------------------------------------------------------------
// Kernel 5: per-edge exp + segment sum into denom[tar] (ebuf -> exp_edge)
// ---------------------------------------------------------------------------
__global__ void __launch_bounds__(256)
gat_edge_exp(const long long* __restrict__ ei, float* __restrict__ ebuf,
             const float* __restrict__ m, float* __restrict__ denom, int n_edges) {
    const int e = blockIdx.x * 256 + threadIdx.x;
    if (e >= n_edges) return;
    const long long t = ei[e];
    const float ee = expf(ebuf[e] - m[t]);
    ebuf[e] = ee;
    atomicAdd(&denom[t], ee);
}

// ---------------------------------------------------------------------------
// Kernel 6: out[n,c] = (exp_self[n]/denom[n]) * feature[n,c]   (initializes out)
// ---------------------------------------------------------------------------
__global__ void __launch_bounds__(256)
gat_self_out(const float* __restrict__ feature, const float* __restrict__ exps,
             const float* __restrict__ denom, float* __restrict__ out, int n_nodes) {
    const int idx = blockIdx.x * 256 + threadIdx.x;
    if (idx >= n_nodes * OUT_CH) return;
    const int n = idx >> 6;           // /64
    out[idx] = (exps[n] / denom[n]) * feature[idx];
}

// ---------------------------------------------------------------------------
// Kernel 7: weighted gather-scatter: out[tar] += alpha * feature[src]
// wave per edge; lane covers channels c and c+32; f32 HW atomics.
// ---------------------------------------------------------------------------
__global__ void __launch_bounds__(256)
gat_edge_agg(const long long* __restrict__ ei, const float* __restrict__ feature,
             const float* __restrict__ ebuf, const float* __restrict__ denom,
             float* __restrict__ out, int n_edges) {
    const int e = blockIdx.x * 8 + (threadIdx.x >> 5);
    const int lane = threadIdx.x & 31;
    if (e >= n_edges) return;
    const long long t = ei[e];
    const long long s = ei[n_edges + e];
    const float alpha = ebuf[e] / denom[t];
    const float* __restrict__ fs = feature + (size_t)s * OUT_CH;
    float* __restrict__ ot = out + (size_t)t * OUT_CH;
    atomicAdd(&ot[lane],      alpha * fs[lane]);
    atomicAdd(&ot[lane + 32], alpha * fs[lane + 32]);
}

// ---------------------------------------------------------------------------
extern "C" void kernel_launch(void* const* d_in, const int* in_sizes, int n_in,
                              void* d_out, int out_size, void* d_ws, size_t ws_size,
                              hipStream_t stream) {
    const float*     x   = (const float*)d_in[0];      // [N, 256]
    const long long* ei  = (const long long*)d_in[1];  // [2, E] int64
    const float*     W   = (const float*)d_in[2];      // [64, 256]
    const float*     b   = (const float*)d_in[3];      // [64]
    const float*     att = (const float*)d_in[4];      // [64, 2]
    float*           out = (float*)d_out;              // [N, 64]

    const int N = in_sizes[0] / IN_CH;   // 50000
    const int E = in_sizes[1] / 2;       // 800000

    // workspace carve-out (all float32, ~17 MB total)
    float* ws      = (float*)d_ws;
    float* feature = ws;                               // N*64
    float* e_pre   = feature + (size_t)N * OUT_CH;     // N*2
    float* m       = e_pre   + (size_t)N * 2;          // N
    float* denom   = m       + N;                      // N
    float* exps    = denom   + N;                      // N
    float* ebuf    = exps    + N;                      // E

    gat_gemm_wmma <<<(N + 15) / 16,           128, 0, stream>>>(x, W, b, feature, N);
    gat_epre      <<<(N + 7) / 8,             256, 0, stream>>>(feature, att, e_pre, m, N);
    gat_edge_logit<<<(E + 255) / 256,         256, 0, stream>>>(ei, e_pre, ebuf, m, E);
    gat_denom_init<<<(N + 255) / 256,         256, 0, stream>>>(e_pre, m, denom, exps, N);
    gat_edge_exp  <<<(E + 255) / 256,         256, 0, stream>>>(ei, ebuf, m, denom, E);
    gat_self_out  <<<(N * OUT_CH + 255) / 256,256, 0, stream>>>(feature, exps, denom, out, N);
    gat_edge_agg  <<<(E + 7) / 8,             256, 0, stream>>>(ei, feature, ebuf, denom, out, E);
}